// KA_BSpline_Core_8761733284002
// MI455X (gfx1250) — compile-verified
//
#include <hip/hip_runtime.h>

typedef __attribute__((ext_vector_type(2))) float v2f;
typedef __attribute__((ext_vector_type(8))) float v8f;

#define QDIM   12
#define NCTRL  25
#define DIM    256
#define WAVES  4                 // waves per block -> 128 threads
#define WROW   260               // sW row stride (floats): 260 % 64 == 4 -> bank-conflict-free
#define CROW   68                // chunk row stride (floats): 68 % 64 == 4 -> conflict-free
#define CHUNK_COLS   64          // K-columns per async chunk
#define NCHUNK       (DIM / CHUNK_COLS)   // 4
#define CHUNK_FLOATS (16 * CROW)          // 1088 floats = 4.25 KB per chunk buffer

// knots: [0,0,0,0, 1/22..21/22, 1,1,1,1]  => t(i) = clamp((i-3)/22, 0, 1)
__device__ __forceinline__ float knotv(int i) {
    float v = (float)(i - 3) * (1.0f / 22.0f);
    return fminf(fmaxf(v, 0.0f), 1.0f);
}

// DMA one 16x64-float chunk of the wave's tile into LDS (8 x 512B coalesced bursts).
// Instruction g copies rows {2g, 2g+1}: lanes 0-15 -> row 2g, lanes 16-31 -> row 2g+1,
// each lane moving 16 contiguous bytes. LDS rows are padded to CROW floats.
__device__ __forceinline__ void issue_chunk_async(const float* __restrict__ emb,
                                                  unsigned tileByte, int ck,
                                                  unsigned ldsBase, int lane)
{
    unsigned g = tileByte + (unsigned)ck * (CHUNK_COLS * 4u)
               + ((unsigned)(lane >> 4)) * (DIM * 4u)
               + ((unsigned)(lane & 15)) * 16u;
    unsigned l = ldsBase + ((unsigned)(lane >> 4)) * (CROW * 4u)
               + ((unsigned)(lane & 15)) * 16u;
    #pragma unroll
    for (int gi = 0; gi < 8; ++gi) {
        asm volatile("global_load_async_to_lds_b128 %0, %1, %2"
                     :: "v"(l), "v"(g), "s"(emb)
                     : "memory");
        g += 2u * DIM * 4u;      // two global rows per instruction
        l += 2u * CROW * 4u;
    }
}

__launch_bounds__(WAVES * 32)
__global__ void kan_bspline_wmma(const float* __restrict__ emb,
                                 const float* __restrict__ W,
                                 const float* __restrict__ b_inner,
                                 const float* __restrict__ inner_scale,
                                 const float* __restrict__ coeffs,
                                 const float* __restrict__ a,
                                 const float* __restrict__ b_out,
                                 float* __restrict__ out,
                                 int npts)
{
    __shared__ float sW[16 * WROW];                    // W padded to 16 rows, padded stride
    __shared__ float sC[QDIM * NCTRL];                 // spline coefficients
    __shared__ float sT[WAVES * 2 * CHUNK_FLOATS];     // double-buffered chunks per wave

    const int tid  = threadIdx.x;
    const int nthr = WAVES * 32;

    // Stage W (zero-padded rows 12..15) and coeffs into LDS, coalesced.
    for (int idx = tid; idx < 16 * DIM; idx += nthr) {
        int r = idx >> 8, c = idx & 255;
        sW[r * WROW + c] = (r < QDIM) ? W[idx] : 0.0f;
    }
    for (int idx = tid; idx < QDIM * NCTRL; idx += nthr) sC[idx] = coeffs[idx];
    __syncthreads();

    const int lane  = tid & 31;
    const int wv    = tid >> 5;
    const int row16 = lane & 15;             // M row (A) / N col (B,C,D)
    const int colo  = (lane >> 4) << 1;      // K sub-offset: 0 or 2
    const int qc    = (row16 < QDIM) ? row16 : (QDIM - 1);

    const float scale = inner_scale[0];
    const float bq    = b_inner[qc];
    const float aq    = (row16 < QDIM) ? a[qc] : 0.0f;   // masks padded q=12..15
    const float bo    = b_out[0];

    const int tiles   = (npts + 15) >> 4;
    const int wstride = gridDim.x * WAVES;

    float*   buf0 = sT + (wv * 2 + 0) * CHUNK_FLOATS;
    float*   buf1 = sT + (wv * 2 + 1) * CHUNK_FLOATS;
    float*   bufs[2]   = { buf0, buf1 };
    unsigned bufLds[2] = { (unsigned)(size_t)(void*)buf0,
                           (unsigned)(size_t)(void*)buf1 };

    const float* wbase = sW + row16 * WROW + colo;

    int tile = blockIdx.x * WAVES + wv;
    if (tile < tiles)
        issue_chunk_async(emb, (unsigned)tile * (16u * DIM * 4u), 0, bufLds[0], lane);

    for (; tile < tiles; tile += wstride) {
        v8f acc0 = {};
        v8f acc1 = {};
        #pragma unroll
        for (int ck = 0; ck < NCHUNK; ++ck) {
            // Prefetch the next chunk (possibly first chunk of this wave's next tile).
            const int ntile = (ck < NCHUNK - 1) ? tile : (tile + wstride);
            const int nck   = (ck < NCHUNK - 1) ? (ck + 1) : 0;
            if (ntile < tiles) {
                issue_chunk_async(emb, (unsigned)ntile * (16u * DIM * 4u), nck,
                                  bufLds[(ck + 1) & 1], lane);
                // 8 outstanding allowed = the chunk just issued; in-order completion
                // guarantees the current chunk's 8 transfers have landed.
                asm volatile("s_wait_asynccnt 0x8" ::: "memory");
            } else {
                asm volatile("s_wait_asynccnt 0x0" ::: "memory");
            }

            const float* ap = bufs[ck & 1] + row16 * CROW + colo;
            const float* bp = wbase + ck * CHUNK_COLS;
            #pragma unroll
            for (int i = 0; i < CHUNK_COLS / 4; i += 2) {
                v2f av0 = *(const v2f*)(ap + 4 * i);
                v2f bv0 = *(const v2f*)(bp + 4 * i);
                v2f av1 = *(const v2f*)(ap + 4 * i + 4);
                v2f bv1 = *(const v2f*)(bp + 4 * i + 4);
                acc0 = __builtin_amdgcn_wmma_f32_16x16x4_f32(false, av0, false, bv0,
                                                             (short)0, acc0, false, false);
                acc1 = __builtin_amdgcn_wmma_f32_16x16x4_f32(false, av1, false, bv1,
                                                             (short)0, acc1, false, false);
            }
        }
        v8f acc = acc0 + acc1;

        // C/D layout: VGPR j, lanes 0-15 -> point tile*16+j, lanes 16-31 -> +8.
        const int pbase = tile * 16 + (lane >> 4) * 8;
        #pragma unroll
        for (int j = 0; j < 8; ++j) {
            float u  = acc[j] + bq;
            float sv = 0.5f * (tanhf(scale * u) + 1.0f);   // in (0,1)

            int kk = (int)(sv * 22.0f);
            kk = kk < 0 ? 0 : (kk > 21 ? 21 : kk);
            const int span = kk + 3;

            float Nb[4] = {1.0f, 0.0f, 0.0f, 0.0f};
            float lft[4], rgt[4];
            #pragma unroll
            for (int jj = 1; jj <= 3; ++jj) {
                lft[jj] = sv - knotv(span + 1 - jj);
                rgt[jj] = knotv(span + jj) - sv;
                float saved = 0.0f;
                #pragma unroll
                for (int r = 0; r < jj; ++r) {
                    float temp = Nb[r] / (rgt[r + 1] + lft[jj - r]);
                    Nb[r] = saved + rgt[r + 1] * temp;
                    saved = lft[jj - r] * temp;
                }
                Nb[jj] = saved;
            }

            const float* cp = sC + qc * NCTRL + (span - 3);
            float phi = Nb[0] * cp[0] + Nb[1] * cp[1] + Nb[2] * cp[2] + Nb[3] * cp[3];
            float contrib = phi * aq;

            #pragma unroll
            for (int m = 1; m < 16; m <<= 1)
                contrib += __shfl_xor(contrib, m, 32);

            if (row16 == 0)
                out[pbase + j] = contrib + bo;
        }
    }
}

extern "C" void kernel_launch(void* const* d_in, const int* in_sizes, int n_in,
                              void* d_out, int out_size, void* d_ws, size_t ws_size,
                              hipStream_t stream) {
    const float* emb         = (const float*)d_in[0];
    const float* W           = (const float*)d_in[1];
    const float* b_inner     = (const float*)d_in[2];
    const float* inner_scale = (const float*)d_in[3];
    const float* coeffs      = (const float*)d_in[4];
    const float* a           = (const float*)d_in[5];
    const float* b_out       = (const float*)d_in[6];
    float* out               = (float*)d_out;

    const int npts  = in_sizes[0] / DIM;            // 500000
    const int tiles = (npts + 15) / 16;             // 31250
    int blocks = (tiles + WAVES - 1) / WAVES;
    if (blocks > 2048) blocks = 2048;               // grid-stride

    hipLaunchKernelGGL(kan_bspline_wmma, dim3(blocks), dim3(WAVES * 32), 0, stream,
                       emb, W, b_inner, inner_scale, coeffs, a, b_out, out, npts);
}